// DKVB_86242943304021
// MI455X (gfx1250) — compile-verified
//
#include <hip/hip_runtime.h>
#include <stdint.h>

typedef __attribute__((ext_vector_type(2))) float v2f;
typedef __attribute__((ext_vector_type(8))) float v8f;

#define NCB     256   // codebooks
#define NCODES  4096  // codes per codebook
#define CBD     8     // codebook dim
#define NQ      128   // B*H*W queries per codebook
#define EMBD    2048
#define CHUNK   512   // codes staged in LDS per buffer
#define KPAD    12    // padded row stride (floats): 48B -> 16B-aligned rows, conflict-free frags
#define TILES_PER_CHUNK (CHUNK / 16)
#define NCHUNK  (NCODES / CHUNK)

// CDNA5 async global->LDS copy (VGLOBAL GLOBAL_LOAD_ASYNC_TO_LDS_B128, ASYNCcnt-tracked).
// VDST = LDS byte offset (low 32 bits of generic LDS pointer), VADDR = 64-bit global addr.
__device__ __forceinline__ void async_copy_b128(const float* gsrc, float* lds_dst) {
    uint32_t loff = (uint32_t)(uintptr_t)lds_dst;   // generic LDS ptr low 32 bits = LDS offset
    asm volatile("global_load_async_to_lds_b128 %0, %1, off"
                 :
                 : "v"(loff), "v"(gsrc)
                 : "memory");
}
__device__ __forceinline__ void wait_async_zero() {
    asm volatile("s_wait_asynccnt 0x0" ::: "memory");
}

// Issue one chunk (512 codes x 8 floats = 1024 float4s) into an LDS buffer.
__device__ __forceinline__ void issue_chunk_load(const float* __restrict__ keysC,
                                                 float (*skbuf)[KPAD],
                                                 int base, int tid) {
    const float* src = keysC + (size_t)base * CBD;
    #pragma unroll
    for (int i = 0; i < 4; ++i) {
        int f = tid + i * 256;            // float4 id 0..1023
        int code = f >> 1, k0 = (f & 1) * 4;
        async_copy_b128(src + (size_t)f * 4, &skbuf[code][k0]);  // 16B, 16B-aligned dest
    }
}

__global__ __launch_bounds__(256) void dkvb_argmin_gather(
    const float* __restrict__ emb,     // (8, 2048, 16)
    const float* __restrict__ keys,    // (256, 4096, 8)
    const float* __restrict__ values,  // (256, 4096, 8)
    float* __restrict__ out)           // (8, 2048, 16)
{
    __shared__ float sX[NQ][CBD];            // 4 KB
    __shared__ float sK[2][CHUNK][KPAD];     // 48 KB (double-buffered keys)
    __shared__ float sNorm[2][CHUNK];        // 4 KB

    const int c    = blockIdx.x;
    const int tid  = threadIdx.x;
    const int lane = tid & 31;
    const int wave = tid >> 5;
    const int n    = lane & 15;              // column within 16-wide tile
    const int kh   = lane >> 4;              // K half-select (ISA A/B frag layout)
    const int m0   = wave * 16;

    const float* keysC = keys   + (size_t)c * NCODES * CBD;
    const float* valsC = values + (size_t)c * NCODES * CBD;

    // stage query slice: sX[m][k] = emb[(b*2048 + c*8 + k)*16 + p]
    #pragma unroll
    for (int i = 0; i < 4; ++i) {
        int e = tid + i * 256;
        int m = e >> 3, k = e & 7;
        int b = m >> 4, p = m & 15;
        sX[m][k] = emb[((size_t)b * EMBD + (size_t)c * CBD + k) * 16 + p];
    }

    // prologue: start async copy of chunk 0 while X staging settles
    issue_chunk_load(keysC, sK[0], 0, tid);
    __syncthreads();

    // loop-invariant A fragments (16x4 f32 layout), pre-scaled by -2
    const int mrow = m0 + n;
    v2f a0, a1;
    a0[0] = -2.0f * sX[mrow][2 * kh + 0];
    a0[1] = -2.0f * sX[mrow][2 * kh + 1];
    a1[0] = -2.0f * sX[mrow][2 * kh + 4];
    a1[1] = -2.0f * sX[mrow][2 * kh + 5];

    float best[8];
    int   bidx[8];
    #pragma unroll
    for (int j = 0; j < 8; ++j) { best[j] = 3.402823466e38f; bidx[j] = 0; }

    for (int ch = 0; ch < NCHUNK; ++ch) {
        const int cur  = ch & 1;
        const int base = ch * CHUNK;

        wait_async_zero();                      // chunk `ch` landed in sK[cur]
        __syncthreads();                        // all waves' copies visible; prev buffer free

        if (ch + 1 < NCHUNK)                    // overlap next chunk with this one's compute
            issue_chunk_load(keysC, sK[cur ^ 1], base + CHUNK, tid);

        // per-code squared norms for current chunk
        #pragma unroll
        for (int i = 0; i < 2; ++i) {
            int code = tid + i * 256;
            float s = 0.0f;
            #pragma unroll
            for (int k = 0; k < 8; ++k) { float kv = sK[cur][code][k]; s += kv * kv; }
            sNorm[cur][code] = s;
        }
        __syncthreads();

        // 32 N-tiles; WMMA accumulates -2*x.k from literal-0 C, norm added at compare
        for (int t = 0; t < TILES_PER_CHUNK; ++t) {
            const int row = t * 16 + n;
            const float nv = sNorm[cur][row];

            v2f b0, b1;                          // B frag (4x16): lane->N, VGPR+half->K
            b0[0] = sK[cur][row][2 * kh + 0]; b0[1] = sK[cur][row][2 * kh + 1];
            b1[0] = sK[cur][row][2 * kh + 4]; b1[1] = sK[cur][row][2 * kh + 5];

            v8f acc = {};                        // SRC2 = inline 0, no broadcast movs
            acc = __builtin_amdgcn_wmma_f32_16x16x4_f32(
                      false, a0, false, b0, (short)0, acc, false, false);
            acc = __builtin_amdgcn_wmma_f32_16x16x4_f32(
                      false, a1, false, b1, (short)0, acc, false, false);

            const int cidx = base + t * 16 + n;
            #pragma unroll
            for (int j = 0; j < 8; ++j) {
                float dist = acc[j] + nv;        // ||k||^2 - 2*x.k
                bool lt = dist < best[j];
                best[j] = lt ? dist : best[j];
                bidx[j] = lt ? cidx : bidx[j];
            }
        }
        __syncthreads();                         // readers done before buffer reuse
    }

    // per-row argmin across the 16 lanes of each half, then gather + scatter
    #pragma unroll
    for (int j = 0; j < 8; ++j) {
        float v  = best[j];
        int   bi = bidx[j];
        #pragma unroll
        for (int mask = 8; mask >= 1; mask >>= 1) {
            float ov = __shfl_xor(v, mask, 32);
            int   oi = __shfl_xor(bi, mask, 32);
            if (ov < v || (ov == v && oi < bi)) { v = ov; bi = oi; }
        }
        if (n == 0) {
            int m = m0 + j + 8 * kh;             // D layout: VGPR j -> row j + 8*(lane>=16)
            int b = m >> 4, p = m & 15;
            const float4* vp = (const float4*)(valsC + (size_t)bi * CBD);
            float4 v0 = vp[0], v1 = vp[1];
            float* op = out + ((size_t)b * EMBD + (size_t)c * CBD) * 16 + p;
            op[0]   = v0.x; op[16]  = v0.y; op[32]  = v0.z; op[48]  = v0.w;
            op[64]  = v1.x; op[80]  = v1.y; op[96]  = v1.z; op[112] = v1.w;
        }
    }
}

extern "C" void kernel_launch(void* const* d_in, const int* in_sizes, int n_in,
                              void* d_out, int out_size, void* d_ws, size_t ws_size,
                              hipStream_t stream) {
    const float* emb    = (const float*)d_in[0];   // (8, 2048, 4, 4)
    const float* keys   = (const float*)d_in[1];   // (256, 4096, 8)
    const float* values = (const float*)d_in[2];   // (256, 4096, 8)
    float* out = (float*)d_out;                    // (8, 2048, 4, 4)

    dim3 grid(NCB), block(256);
    dkvb_argmin_gather<<<grid, block, 0, stream>>>(emb, keys, values, out);
}